// GCNLayer_77309411328169
// MI455X (gfx1250) — compile-verified
//
#include <hip/hip_runtime.h>

// GCN layer for MI455X (gfx1250, wave32).
//   feat = [user_feat @ u_w ; item_feat @ v_w]   (fp32 WMMA 16x16x4, W staged in LDS
//                                                 pre-swizzled into fragment order)
//   out[r] += vals[e] * feat[cols[e]]            (row-sorted COO, segmented flush:
//                                                 native f32 atomics at chunk edges,
//                                                 plain b64 stores for interior rows)
//   out = relu(out)

typedef __attribute__((ext_vector_type(2))) float v2f;
typedef __attribute__((ext_vector_type(8))) float v8f;

#define IN_DIM   64
#define OUT_DIM  64
#define N_USERS  100000
#define N_ITEMS  200000
#define N_NODES  300000
#define N_EDGES  4800000
#define EDGES_PER_WAVE 256   // 4.8M / 256 = 18750 waves, no tail

// Native fp32 atomic add (GLOBAL_ATOMIC_ADD_F32, no return -> STOREcnt),
// device scope so partial sums from different WGPs combine in L2.
__device__ __forceinline__ void atomic_add_f32_dev(float* p, float v) {
  asm volatile("global_atomic_add_f32 %0, %1, off scope:SCOPE_DEV"
               : : "v"(p), "v"(v) : "memory");
}

// ---------------------------------------------------------------------------
// Zero-fill d_out (accumulator) — harness poisons buffers, we must init.
// ---------------------------------------------------------------------------
__global__ void gcn_zero(float4* __restrict__ p, int n4) {
  for (int i = blockIdx.x * blockDim.x + threadIdx.x; i < n4;
       i += gridDim.x * blockDim.x)
    p[i] = make_float4(0.f, 0.f, 0.f, 0.f);
}

// ---------------------------------------------------------------------------
// Dense projection Y[nrows,64] = X[nrows,64] @ W[64,64] using
// V_WMMA_F32_16X16X4_F32. One wave owns a 16-row x 64-col output tile:
// 4 accumulators (v8f), K swept in steps of 4 (16 iters x 4 WMMAs).
//
// W is staged once per block into LDS *pre-swizzled into B-fragment order*:
// pair index p = kb*128 + khalf*64 + n   (kb = K/4, khalf = lane>>4, n = col)
//   sWs[2p+0] = W[4*kb +     khalf][n]   (B VGPR0 element)
//   sWs[2p+1] = W[4*kb + 2 + khalf][n]   (B VGPR1 element)
// so each lane's B fragment is one aligned ds_load_b64 — no repacking movs.
//
// VGPR layouts (ISA 7.12.2):
//   A 16x4 f32: lanes 0-15 -> {K=k0,k0+1}, lanes 16-31 -> {K=k0+2,k0+3}, M=lane&15
//   B  4x16:    b[j], lane l: K = k0 + 2*j + (l>>4), N = l&15   (row-striped)
//   C/D 16x16:  VGPR i, lanes 0-15 -> M=i, lanes 16-31 -> M=i+8, N = lane&15
// nrows is a multiple of 16, so EXEC is all-1s in every WMMA wave.
// ---------------------------------------------------------------------------
__global__ void gcn_proj_wmma(const float* __restrict__ X,
                              const float* __restrict__ W,
                              float* __restrict__ Y, int nrows) {
  __shared__ float sWs[IN_DIM * OUT_DIM];   // 16 KB, fragment-ordered pairs

  // Swizzled staging: 2048 pairs total; each of 256 threads builds 8
  // consecutive pairs (they never straddle a kb/khalf boundary since both
  // change at multiples of 8). Writes are 16 contiguous floats per thread.
  {
    const int t     = threadIdx.x;
    const int p0    = t * 8;
    const int kb    = p0 >> 7;
    const int rem   = p0 & 127;
    const int khalf = rem >> 6;
    const int n0    = rem & 63;
    const float* rA = W + (size_t)(4 * kb + 0 + khalf) * OUT_DIM + n0;
    const float* rB = W + (size_t)(4 * kb + 2 + khalf) * OUT_DIM + n0;
    float*       d  = sWs + 2 * p0;
#pragma unroll
    for (int i = 0; i < 8; i += 2) {
      *(float4*)(d + 2 * i) = make_float4(rA[i], rB[i], rA[i + 1], rB[i + 1]);
    }
  }
  __syncthreads();

  const int lane  = threadIdx.x & 31;
  const int wave  = threadIdx.x >> 5;
  const int tile  = blockIdx.x * (blockDim.x >> 5) + wave;
  const int row0  = tile * 16;
  if (row0 >= nrows) return;                 // wave-uniform exit

  const int m     = lane & 15;               // M (A) / N (B,C) index
  const int khalf = lane >> 4;               // 0 or 1

  v8f acc[4];
#pragma unroll
  for (int nt = 0; nt < 4; ++nt) acc[nt] = (v8f){};

  const float* xrow = X + (size_t)(row0 + m) * IN_DIM;

#pragma unroll
  for (int k0 = 0; k0 < IN_DIM; k0 += 4) {
    // A fragment: contiguous 8B per lane (global_load_b64).
    const float* ap = xrow + k0 + khalf * 2;
    v2f a;
    a.x = ap[0];
    a.y = ap[1];

    const int kb = k0 >> 2;
#pragma unroll
    for (int nt = 0; nt < 4; ++nt) {
      // One aligned ds_load_b64 per fragment.
      v2f b = *(const v2f*)(sWs + 2 * (kb * 128 + khalf * 64 + nt * 16 + m));
      // (neg_a, A, neg_b, B, c_mod, C, reuse_a, reuse_b)
      acc[nt] = __builtin_amdgcn_wmma_f32_16x16x4_f32(
          false, a, false, b, (short)0, acc[nt], false, false);
    }
  }

  // Scatter D back: row = row0 + i + 8*khalf, col = nt*16 + m.
#pragma unroll
  for (int nt = 0; nt < 4; ++nt) {
#pragma unroll
    for (int i = 0; i < 8; ++i) {
      Y[(size_t)(row0 + i + 8 * khalf) * OUT_DIM + nt * 16 + m] = acc[nt][i];
    }
  }
}

// ---------------------------------------------------------------------------
// Row-sorted COO SpMM. One wave owns exactly 256 contiguous edges; each lane
// owns 2 of the 64 feature dims (8B coalesced gather per edge).
//
// Segmented flush policy (rows are globally sorted):
//   - first segment of the chunk and the final segment may straddle chunk
//     boundaries -> flush with native f32 atomics;
//   - every other segment is wholly owned by this wave -> plain b64 store.
// Edge metadata is single-use streaming -> non-temporal loads so the 77 MB
// feat table stays resident in the 192 MB L2 across its ~16x reuse.
// Prefetch runs once per 16-edge group (64B of each metadata stream).
// ---------------------------------------------------------------------------
__global__ void gcn_spmm(const float* __restrict__ vals,
                         const int*   __restrict__ rows,
                         const int*   __restrict__ cols,
                         const float* __restrict__ feat,
                         float*       __restrict__ out,
                         int nEdges) {
  const int  lane = threadIdx.x & 31;
  const int  wave = (blockIdx.x * blockDim.x + threadIdx.x) >> 5;
  const long e0   = (long)wave * EDGES_PER_WAVE;
  if (e0 >= nEdges) return;                  // wave-uniform exit

  const int d0 = lane * 2;

  float ax = 0.0f, ay = 0.0f;
  int  curRow   = __builtin_nontemporal_load(&rows[e0]);
  bool firstSeg = true;

  for (int g = 0; g < EDGES_PER_WAVE / 16; ++g) {
    const long base = e0 + (long)g * 16;
    if (g + 1 < EDGES_PER_WAVE / 16) {       // pull next group's metadata
      __builtin_prefetch(&rows[base + 16], 0, 0);
      __builtin_prefetch(&cols[base + 16], 0, 0);
      __builtin_prefetch(&vals[base + 16], 0, 0);
    }
    for (int i = 0; i < 16; ++i) {
      const long  e = base + i;
      const int   r = __builtin_nontemporal_load(&rows[e]);
      const int   c = __builtin_nontemporal_load(&cols[e]);
      const float v = __builtin_nontemporal_load(&vals[e]);

      if (r != curRow) {                     // wave-uniform branch
        float* dst = out + (size_t)curRow * OUT_DIM + d0;
        if (firstSeg) {
          atomic_add_f32_dev(dst + 0, ax);
          atomic_add_f32_dev(dst + 1, ay);
          firstSeg = false;
        } else {
          *(float2*)dst = make_float2(ax, ay);  // exclusive interior row
        }
        ax = 0.0f;
        ay = 0.0f;
        curRow = r;
      }
      const float* fp = feat + (size_t)c * OUT_DIM + d0;
      ax = fmaf(v, fp[0], ax);
      ay = fmaf(v, fp[1], ay);
    }
  }
  // Final segment may straddle the chunk boundary -> atomics.
  float* dst = out + (size_t)curRow * OUT_DIM + d0;
  atomic_add_f32_dev(dst + 0, ax);
  atomic_add_f32_dev(dst + 1, ay);
}

// ---------------------------------------------------------------------------
// In-place ReLU after all atomic accumulation completed (separate launch).
// ---------------------------------------------------------------------------
__global__ void gcn_relu(float4* __restrict__ p, int n4) {
  for (int i = blockIdx.x * blockDim.x + threadIdx.x; i < n4;
       i += gridDim.x * blockDim.x) {
    float4 x = p[i];
    x.x = fmaxf(x.x, 0.f);
    x.y = fmaxf(x.y, 0.f);
    x.z = fmaxf(x.z, 0.f);
    x.w = fmaxf(x.w, 0.f);
    p[i] = x;
  }
}

// ---------------------------------------------------------------------------
extern "C" void kernel_launch(void* const* d_in, const int* in_sizes, int n_in,
                              void* d_out, int out_size, void* d_ws, size_t ws_size,
                              hipStream_t stream) {
  const float* user_feat = (const float*)d_in[0];   // [100000,64]
  const float* item_feat = (const float*)d_in[1];   // [200000,64]
  const float* u_w       = (const float*)d_in[2];   // [64,64]
  const float* v_w       = (const float*)d_in[3];   // [64,64]
  const float* adj_vals  = (const float*)d_in[4];   // [E]
  const int*   adj_rows  = (const int*)  d_in[5];   // [E] (harness: integer -> int32)
  const int*   adj_cols  = (const int*)  d_in[6];   // [E]

  float* out  = (float*)d_out;                      // [300000,64]
  float* feat = (float*)d_ws;                       // [300000,64] scratch (76.8 MB)

  // 1) zero accumulator
  {
    const int n4 = (N_NODES * OUT_DIM) / 4;
    gcn_zero<<<2048, 256, 0, stream>>>((float4*)out, n4);
  }

  // 2) projections (fp32 WMMA). 8 waves (tiles) per 256-thread block.
  {
    const int tilesU  = N_USERS / 16;                       // 6250
    const int blocksU = (tilesU + 7) / 8;
    gcn_proj_wmma<<<blocksU, 256, 0, stream>>>(user_feat, u_w, feat, N_USERS);

    const int tilesI  = N_ITEMS / 16;                       // 12500
    const int blocksI = (tilesI + 7) / 8;
    gcn_proj_wmma<<<blocksI, 256, 0, stream>>>(item_feat, v_w,
                                               feat + (size_t)N_USERS * OUT_DIM,
                                               N_ITEMS);
  }

  // 3) segmented SpMM: 18750 waves x 256 edges, 8 waves per block.
  {
    const int nWaves  = N_EDGES / EDGES_PER_WAVE;                     // 18750
    const int nBlocks = (nWaves + 7) / 8;                             // 2344
    gcn_spmm<<<nBlocks, 256, 0, stream>>>(adj_vals, adj_rows, adj_cols,
                                          feat, out, N_EDGES);
  }

  // 4) ReLU
  {
    const int n4 = (N_NODES * OUT_DIM) / 4;
    gcn_relu<<<2048, 256, 0, stream>>>((float4*)out, n4);
  }
}